// Attention_89730456748201
// MI455X (gfx1250) — compile-verified
//
#include <hip/hip_runtime.h>
#include <hip/hip_bf16.h>
#include <math.h>

typedef float v2f __attribute__((ext_vector_type(2)));
typedef float v8f __attribute__((ext_vector_type(8)));

#define NB 1024   // batches (stocks)
#define NT 1024   // context length
#define ND 128    // feature dim

// ---------------------------------------------------------------------------
// Kernel 1: q[m,n] = sum_k query[m,k] * W_in[n,k]   (M=1024, N=128, K=128)
// One wave per 16x16 output tile, V_WMMA_F32_16X16X4_F32, 32 K-steps.
// A-matrix lane layout (ISA 7.12.2, 32-bit 16x4): lane%16 = M row,
// (lane/16)*2 = K pair base; 2 VGPRs hold K, K+1 -> contiguous float2 load.
// ---------------------------------------------------------------------------
__global__ void __launch_bounds__(128)
qproj_wmma_kernel(const float* __restrict__ query,
                  const float* __restrict__ Win,
                  float* __restrict__ qws) {
    const int lane = threadIdx.x & 31;
    const int wave = threadIdx.x >> 5;
    const int tile = blockIdx.x * 4 + wave;   // 512 tiles total
    const int m0 = (tile >> 3) * 16;          // 64 M-tiles
    const int n0 = (tile & 7) * 16;           // 8 N-tiles
    const int mn = lane & 15;
    const int kl = (lane >> 4) * 2;

    const float* arow = query + (size_t)(m0 + mn) * ND + kl;  // A[m, k..k+1]
    const float* brow = Win   + (size_t)(n0 + mn) * ND + kl;  // B[k,n]=W_in[n,k]

    v8f c = {};
    #pragma unroll 8
    for (int k0 = 0; k0 < ND; k0 += 4) {
        v2f a = *(const v2f*)(arow + k0);
        v2f b = *(const v2f*)(brow + k0);
        c = __builtin_amdgcn_wmma_f32_16x16x4_f32(
                /*neg_a=*/false, a, /*neg_b=*/false, b,
                /*c_mod=*/(short)0, c, /*reuse_a=*/false, /*reuse_b=*/false);
    }
    const int half = lane >> 4;   // C/D layout: VGPR r -> M = r + 8*half
    #pragma unroll
    for (int r = 0; r < 8; ++r)
        qws[(size_t)(m0 + r + 8 * half) * ND + n0 + mn] = c[r];
}

// ---------------------------------------------------------------------------
// Kernel 2: per-batch attention streaming (bandwidth-bound part).
// 1 block / batch, 256 threads = 8 waves. Each wave streams 128 context rows
// with coalesced float4 loads (lane*16B covers the 512B row).
// Pass 1: scores -> LDS; block softmax in LDS; weights -> d_out.
// Pass 2: decay-weighted accumulation, combined via ds_add_f32 LDS atomics.
// ---------------------------------------------------------------------------
__global__ void __launch_bounds__(256)
attn_stream_kernel(const float* __restrict__ ctx,
                   const float* __restrict__ qws,
                   const float* __restrict__ ae_p,
                   const float* __restrict__ ab_p,
                   float* __restrict__ weights_out,
                   float* __restrict__ comb) {
    const int b    = blockIdx.x;
    const int tid  = threadIdx.x;
    const int lane = tid & 31;
    const int wave = tid >> 5;

    __shared__ float s[NT];        // scores -> exp -> weights
    __shared__ float red[256];     // block reduction scratch
    __shared__ float mixacc[ND];   // mix_sum accumulator

    const float* cb = ctx + (size_t)b * NT * ND;
    const float  ae = ae_p[b];
    const float  ab = ab_p[b];

    // each lane holds 4 features of q[b]
    const float4 q4 = *(const float4*)(qws + (size_t)b * ND + lane * 4);

    // ---- pass 1: scores[t] = q . context[t] ----
    const int t0 = wave * 128, t1 = t0 + 128;
    for (int t = t0; t < t1; ++t) {
        const float* row = cb + (size_t)t * ND + lane * 4;
        if (t + 8 < t1) __builtin_prefetch(row + 8 * ND, 0, 0);
        float4 c4 = *(const float4*)row;
        float p = q4.x * c4.x + q4.y * c4.y + q4.z * c4.z + q4.w * c4.w;
        #pragma unroll
        for (int off = 16; off > 0; off >>= 1) p += __shfl_xor(p, off, 32);
        if (lane == 0) s[t] = p;
    }
    __syncthreads();

    // ---- softmax over s[0..NT) ----
    float m = -3.0e38f;
    for (int i = tid; i < NT; i += 256) m = fmaxf(m, s[i]);
    red[tid] = m; __syncthreads();
    #pragma unroll
    for (int st = 128; st > 0; st >>= 1) {
        if (tid < st) red[tid] = fmaxf(red[tid], red[tid + st]);
        __syncthreads();
    }
    m = red[0];
    __syncthreads();

    float sum = 0.f;
    for (int i = tid; i < NT; i += 256) {
        float e = __expf(s[i] - m);
        s[i] = e;
        sum += e;
    }
    red[tid] = sum; __syncthreads();
    #pragma unroll
    for (int st = 128; st > 0; st >>= 1) {
        if (tid < st) red[tid] += red[tid + st];
        __syncthreads();
    }
    const float inv = 1.0f / red[0];

    float* wout = weights_out + (size_t)b * NT;
    for (int i = tid; i < NT; i += 256) {
        float w = s[i] * inv;
        s[i] = w;
        wout[i] = w;
    }
    if (tid < ND) mixacc[tid] = 0.f;
    __syncthreads();

    // ---- pass 2: mix_sum[d] = sum_t (w*c + relu(ae * w*c * exp(-ab*(T-1-t)))) ----
    float a0 = 0.f, a1 = 0.f, a2 = 0.f, a3 = 0.f;
    for (int t = t0; t < t1; ++t) {
        const float w  = s[t];
        const float bt = __expf(-ab * (float)(NT - 1 - t));
        const float g  = ae * bt;
        const float* row = cb + (size_t)t * ND + lane * 4;
        if (t + 8 < t1) __builtin_prefetch(row + 8 * ND, 0, 0);
        float4 c4 = *(const float4*)row;
        float m0v = w * c4.x, m1v = w * c4.y, m2v = w * c4.z, m3v = w * c4.w;
        a0 += m0v + fmaxf(g * m0v, 0.f);
        a1 += m1v + fmaxf(g * m1v, 0.f);
        a2 += m2v + fmaxf(g * m2v, 0.f);
        a3 += m3v + fmaxf(g * m3v, 0.f);
    }
    atomicAdd(&mixacc[lane * 4 + 0], a0);   // ds_add_f32
    atomicAdd(&mixacc[lane * 4 + 1], a1);
    atomicAdd(&mixacc[lane * 4 + 2], a2);
    atomicAdd(&mixacc[lane * 4 + 3], a3);
    __syncthreads();

    // combined = [mix_sum, q]  -> [NB, 2*ND]
    if (tid < ND) {
        comb[(size_t)b * 2 * ND + tid]      = mixacc[tid];
        comb[(size_t)b * 2 * ND + ND + tid] = qws[(size_t)b * ND + tid];
    }
}

// ---------------------------------------------------------------------------
// Kernel 3: out[m,n] = tanh( sum_k comb[m,k] * W_out[n,k] )  (M=1024,N=128,K=256)
// ---------------------------------------------------------------------------
__global__ void __launch_bounds__(128)
outproj_wmma_kernel(const float* __restrict__ comb,
                    const float* __restrict__ Wout,
                    float* __restrict__ out) {
    const int lane = threadIdx.x & 31;
    const int wave = threadIdx.x >> 5;
    const int tile = blockIdx.x * 4 + wave;   // 512 tiles
    const int m0 = (tile >> 3) * 16;
    const int n0 = (tile & 7) * 16;
    const int mn = lane & 15;
    const int kl = (lane >> 4) * 2;
    const int K  = 2 * ND;

    const float* arow = comb + (size_t)(m0 + mn) * K + kl;   // combined row
    const float* brow = Wout + (size_t)(n0 + mn) * K + kl;   // B[k,n]=W_out[n,k]

    v8f c = {};
    #pragma unroll 8
    for (int k0 = 0; k0 < K; k0 += 4) {
        v2f a = *(const v2f*)(arow + k0);
        v2f b = *(const v2f*)(brow + k0);
        c = __builtin_amdgcn_wmma_f32_16x16x4_f32(
                false, a, false, b, (short)0, c, false, false);
    }
    const int half = lane >> 4;
    #pragma unroll
    for (int r = 0; r < 8; ++r)
        out[(size_t)(m0 + r + 8 * half) * ND + n0 + mn] = tanhf(c[r]);
}

// ---------------------------------------------------------------------------
extern "C" void kernel_launch(void* const* d_in, const int* in_sizes, int n_in,
                              void* d_out, int out_size, void* d_ws, size_t ws_size,
                              hipStream_t stream) {
    const float* query = (const float*)d_in[0];   // [B,1,D]
    const float* ctx   = (const float*)d_in[1];   // [B,T,D]
    const float* Win   = (const float*)d_in[2];   // [D,D]
    const float* Wout  = (const float*)d_in[3];   // [D,2D]
    const float* ae    = (const float*)d_in[4];   // [B]
    const float* ab    = (const float*)d_in[5];   // [B]

    float* out_p     = (float*)d_out;             // out: B*D floats
    float* weights_p = out_p + (size_t)NB * ND;   // weights: B*T floats

    float* qws  = (float*)d_ws;                   // [B, D]
    float* comb = qws + (size_t)NB * ND;          // [B, 2D]

    qproj_wmma_kernel<<<128, 128, 0, stream>>>(query, Win, qws);
    attn_stream_kernel<<<NB, 256, 0, stream>>>(ctx, qws, ae, ab, weights_p, comb);
    outproj_wmma_kernel<<<128, 128, 0, stream>>>(comb, Wout, out_p);
}